// Encoder_69243462746830
// MI455X (gfx1250) — compile-verified
//
#include <hip/hip_runtime.h>
#include <math.h>

// ---------------------------------------------------------------------------
// 2-layer GCN encoder for MI455X (gfx1250, wave32).
// Dense GEMMs use native fp32 WMMA (V_WMMA_F32_16X16X4_F32): the workload is
// edge-scatter / L2-bandwidth bound (feature matrices, 51 MB, live in the
// 192 MB L2), so fp32 precision is free — no reason to downcast.
// ---------------------------------------------------------------------------

typedef __attribute__((ext_vector_type(2))) float v2f;
typedef __attribute__((ext_vector_type(8))) float v8f;

static constexpr int HID    = 256;
static constexpr int IN_DIM = 64;

// ---- degree / normalization ------------------------------------------------
__global__ void init_deg_kernel(float* __restrict__ deg, int n) {
  int i = blockIdx.x * blockDim.x + threadIdx.x;
  if (i < n) deg[i] = 1.0f;  // self-loop contribution
}

__global__ void accum_deg_kernel(const int* __restrict__ tgt,
                                 float* __restrict__ deg, int e) {
  int i = blockIdx.x * blockDim.x + threadIdx.x;
  if (i < e) atomicAdd(&deg[tgt[i]], 1.0f);
}

__global__ void rsqrt_kernel(float* __restrict__ deg, int n) {
  int i = blockIdx.x * blockDim.x + threadIdx.x;
  if (i < n) deg[i] = rsqrtf(deg[i]);  // deg >= 1 always (self-loops)
}

// ---- simple fills ----------------------------------------------------------
__global__ void zero_kernel(float* __restrict__ p, int n) {
  int i = blockIdx.x * blockDim.x + threadIdx.x;
  if (i < n) p[i] = 0.0f;
}

__global__ void init_out_bias_kernel(float* __restrict__ out,
                                     const float* __restrict__ b, int total) {
  int i = blockIdx.x * blockDim.x + threadIdx.x;
  if (i < total) out[i] = b[i & (HID - 1)];
}

// ---- fp32 WMMA GEMM:  C[M x 256] = A[M x K] @ B[K x 256] -------------------
// One wave32 computes one 16x16 output tile, K stepped by 4 per WMMA.
// A-operand layout (ISA 7.12.2, 32-bit 16x4): lane m=L%16, half h=L/16,
//   vgpr j holds A[m][k0 + 2h + j].
// B-operand layout (4x16): lane n=L%16, vgpr j holds B[k0 + 2h + j][n].
// C/D layout: vgpr r -> row (8h + r), col = L%16.
template <int K>
__global__ void wmma_gemm_kernel(const float* __restrict__ A,
                                 const float* __restrict__ B,
                                 float* __restrict__ C, int M) {
  const int lane = threadIdx.x & 31;
  const int wid  = (blockIdx.x * blockDim.x + threadIdx.x) >> 5;  // wave-uniform
  const int tilesN = HID / 16;                                    // 16
  const int tilesM = (M + 15) / 16;
  if (wid >= tilesM * tilesN) return;

  const int tm = wid / tilesN;
  const int tn = wid % tilesN;
  const int h  = lane >> 4;       // 0 or 1: which K-half this lane holds
  const int ln = lane & 15;

  const int m = tm * 16 + ln;     // A row for this lane
  const int n = tn * 16 + ln;     // B/C column for this lane

  v8f acc = {};
#pragma unroll 4
  for (int k0 = 0; k0 < K; k0 += 4) {
    v2f a, b;
    const float* ap = A + (size_t)m * K + (k0 + 2 * h);
    a.x = ap[0];
    a.y = ap[1];
    b.x = B[(size_t)(k0 + 2 * h) * HID + n];
    b.y = B[(size_t)(k0 + 2 * h + 1) * HID + n];
    // 8 args: (neg_a, A, neg_b, B, c_mod, C, reuse_a, reuse_b)
    acc = __builtin_amdgcn_wmma_f32_16x16x4_f32(
        false, a, false, b, (short)0, acc, false, false);
  }

  float* cp = C + (size_t)(tm * 16 + 8 * h) * HID + n;
#pragma unroll
  for (int r = 0; r < 8; ++r) cp[(size_t)r * HID] = acc[r];
}

// ---- normalized edge scatter: out[t] += H[s] * dinv[s]*dinv[t] -------------
// One wave32 per edge slot; slots [0,E) are real edges, [E, E+N) self-loops.
// 256 floats/row = 8 floats per lane. f32 atomics resolve in L2.
__global__ void scatter_kernel(const float* __restrict__ H,
                               const int* __restrict__ src,
                               const int* __restrict__ tgt,
                               const float* __restrict__ dinv,
                               float* __restrict__ out, int E, int N) {
  const int wid  = (blockIdx.x * blockDim.x + threadIdx.x) >> 5;
  const int lane = threadIdx.x & 31;
  if (wid >= E + N) return;

  int s, t;
  if (wid < E) { s = src[wid]; t = tgt[wid]; }
  else         { s = t = wid - E; }

  const float norm = dinv[s] * dinv[t];
  const float* hp = H + (size_t)s * HID;
  float* op = out + (size_t)t * HID;
#pragma unroll
  for (int j = 0; j < 8; ++j) {
    const int c = lane + 32 * j;
    atomicAdd(op + c, hp[c] * norm);
  }
}

// ---- layer-1 epilogue: X2 = relu(agg + b1 + sinusoidal_pe) -----------------
__global__ void epilogue1_kernel(const float* __restrict__ agg,
                                 const float* __restrict__ b1,
                                 float* __restrict__ X2, int total) {
  int idx = blockIdx.x * blockDim.x + threadIdx.x;
  if (idx >= total) return;
  const int i = idx >> 8;          // node
  const int c = idx & 255;         // channel
  // pe[i, 2k]   = sin(i * exp(-2k * ln(1e4)/256))
  // pe[i, 2k+1] = cos(same angle)  -> exponent index = (c & ~1)
  const float kLn1e4Over256 = 9.210340371976184f / 256.0f;
  const float freq = expf(-(float)(c & ~1) * kLn1e4Over256);
  const float ang  = (float)i * freq;
  const float pe   = (c & 1) ? cosf(ang) : sinf(ang);
  const float v = agg[idx] + b1[c] + pe;
  X2[idx] = v > 0.0f ? v : 0.0f;
}

// ---------------------------------------------------------------------------
extern "C" void kernel_launch(void* const* d_in, const int* in_sizes, int n_in,
                              void* d_out, int out_size, void* d_ws, size_t ws_size,
                              hipStream_t stream) {
  const float* X  = (const float*)d_in[0];   // [N, 64]
  const int*   EI = (const int*)d_in[1];     // [2, E]
  const float* W1 = (const float*)d_in[2];   // [64, 256]
  const float* b1 = (const float*)d_in[3];   // [256]
  const float* W2 = (const float*)d_in[4];   // [256, 256]
  const float* b2 = (const float*)d_in[5];   // [256]
  float* out = (float*)d_out;                // [N, 256]

  const int N = in_sizes[0] / IN_DIM;        // 50000
  const int E = in_sizes[1] / 2;             // 800000
  const int NF = N * HID;                    // 12.8M floats

  // workspace: dinv | bufH | bufAgg  (~103 MB)
  float* dinv   = (float*)d_ws;
  float* bufH   = dinv + (1 << 16);          // 256 KB pad keeps buffers aligned
  float* bufAgg = bufH + (size_t)NF;

  const int* src = EI;
  const int* tgt = EI + E;

  const int T = 256;
  auto cdiv = [](int a, int b) { return (a + b - 1) / b; };

  // 1) symmetric-norm degrees: deg = 1 + count(tgt); dinv = rsqrt(deg)
  init_deg_kernel<<<cdiv(N, T), T, 0, stream>>>(dinv, N);
  accum_deg_kernel<<<cdiv(E, T), T, 0, stream>>>(tgt, dinv, E);
  rsqrt_kernel<<<cdiv(N, T), T, 0, stream>>>(dinv, N);

  // 2) H1 = X @ W1   (fp32 WMMA, K=64)
  {
    const int tiles = cdiv(N, 16) * (HID / 16);
    wmma_gemm_kernel<IN_DIM><<<cdiv(tiles, T / 32), T, 0, stream>>>(X, W1, bufH, N);
  }

  // 3) agg1 = D^-1/2 (A+I) D^-1/2 H1
  zero_kernel<<<cdiv(NF, T), T, 0, stream>>>(bufAgg, NF);
  scatter_kernel<<<cdiv(E + N, T / 32), T, 0, stream>>>(bufH, src, tgt, dinv,
                                                        bufAgg, E, N);

  // 4) X2 = relu(agg1 + b1 + pe)   (overwrite bufH)
  epilogue1_kernel<<<cdiv(NF, T), T, 0, stream>>>(bufAgg, b1, bufH, NF);

  // 5) H2 = X2 @ W2   (fp32 WMMA, K=256; overwrite bufAgg)
  {
    const int tiles = cdiv(N, 16) * (HID / 16);
    wmma_gemm_kernel<HID><<<cdiv(tiles, T / 32), T, 0, stream>>>(bufH, W2, bufAgg, N);
  }

  // 6) out = b2 + D^-1/2 (A+I) D^-1/2 H2
  init_out_bias_kernel<<<cdiv(NF, T), T, 0, stream>>>(out, b2, NF);
  scatter_kernel<<<cdiv(E + N, T / 32), T, 0, stream>>>(bufAgg, src, tgt, dinv,
                                                        out, E, N);
}